// GATq_68839735820549
// MI455X (gfx1250) — compile-verified
//
#include <hip/hip_runtime.h>
#include <math.h>

typedef __attribute__((ext_vector_type(2))) float v2f;
typedef __attribute__((ext_vector_type(8))) float v8f;

#define NFEAT  128
#define NHID   64
#define NHEADS 4
#define NCLASS 40
#define H1     (NHEADS * NHID)   // 256

// ---- monotone float <-> unsigned key (for atomic float max) --------------
__device__ __forceinline__ unsigned fkey(float f) {
    unsigned u = __float_as_uint(f);
    return ((int)u >= 0) ? (u ^ 0x80000000u) : ~u;
}
__device__ __forceinline__ float keyf(unsigned k) {
    unsigned u = ((int)k < 0) ? (k ^ 0x80000000u) : ~k;
    return __uint_as_float(u);
}
__device__ __forceinline__ float lrelu(float v) { return v > 0.f ? v : 0.1f * v; }

// ---- zero-init workspace region ------------------------------------------
__global__ void k_zero(unsigned* __restrict__ p, long long n) {
    long long t = (long long)blockIdx.x * blockDim.x + threadIdx.x;
    if (t < n) p[t] = 0u;
}

// ---- GEMM1: Wh1[N,256] = x[N,128] @ W[head][128,64] (fp32 WMMA) ----------
// grid (N/16, 4), block 128 (4 waves). Wave w handles col tile (blockIdx.y*4+w).
__global__ void k_gemm1(const float* __restrict__ x, const float* __restrict__ W,
                        float* __restrict__ Wh, int N) {
    const int wave  = threadIdx.x >> 5;
    const int lane  = threadIdx.x & 31;
    const int lg    = lane >> 4;       // half-wave group
    const int lm    = lane & 15;
    const int ctile = blockIdx.y * 4 + wave;     // 0..15
    const int m0    = blockIdx.x * 16;
    const int c0    = ctile * 16;
    const int head  = c0 >> 6;
    const int j     = (c0 & 63) + lm;            // column inside head

    const float* __restrict__ A = x + (size_t)(m0 + lm) * NFEAT;
    const float* __restrict__ B = W + (size_t)head * NFEAT * NHID + j;

    v8f acc = {};
#pragma unroll
    for (int k = 0; k < NFEAT; k += 4) {
        const int k0 = k + 2 * lg;
        v2f a, b;
        a.x = A[k0];
        a.y = A[k0 + 1];
        b.x = B[(size_t)k0 * NHID];
        b.y = B[(size_t)(k0 + 1) * NHID];
        acc = __builtin_amdgcn_wmma_f32_16x16x4_f32(false, a, false, b,
                                                    (short)0, acc, false, false);
    }
    float* __restrict__ C = Wh + (size_t)(m0 + 8 * lg) * H1 + c0 + lm;
#pragma unroll
    for (int v = 0; v < 8; ++v) C[(size_t)v * H1] = acc[v];
}

// ---- GEMM2: Wh2[N,40] = h[N,256] @ Wout[256,40] (fp32 WMMA, padded cols) -
// grid (N/16, 3), block 32 (1 wave per 16x16 tile).
__global__ void k_gemm2(const float* __restrict__ h, const float* __restrict__ Wout,
                        float* __restrict__ Wh2, int N) {
    const int lane = threadIdx.x & 31;
    const int lg   = lane >> 4;
    const int lm   = lane & 15;
    const int m0   = blockIdx.x * 16;
    const int c    = blockIdx.y * 16 + lm;

    const float* __restrict__ A = h + (size_t)(m0 + lm) * H1;
    const bool inb = (c < NCLASS);

    v8f acc = {};
#pragma unroll 8
    for (int k = 0; k < H1; k += 4) {
        const int k0 = k + 2 * lg;
        v2f a, b;
        a.x = A[k0];
        a.y = A[k0 + 1];
        b.x = inb ? Wout[(size_t)k0 * NCLASS + c] : 0.f;
        b.y = inb ? Wout[(size_t)(k0 + 1) * NCLASS + c] : 0.f;
        acc = __builtin_amdgcn_wmma_f32_16x16x4_f32(false, a, false, b,
                                                    (short)0, acc, false, false);
    }
    if (inb) {
        float* __restrict__ C = Wh2 + (size_t)(m0 + 8 * lg) * NCLASS + c;
#pragma unroll
        for (int v = 0; v < 8; ++v) C[(size_t)v * NCLASS] = acc[v];
    }
}

// ---- per-node attention scores, layer 1: sr/sc[N,4] ----------------------
__global__ void k_scores1(const float* __restrict__ Wh, const float* __restrict__ a,
                          float* __restrict__ sr, float* __restrict__ sc, int N) {
    long long t = (long long)blockIdx.x * blockDim.x + threadIdx.x;
    if (t >= (long long)N * NHEADS) return;
    const int n = (int)(t >> 2), hh = (int)(t & 3);
    const float* __restrict__ w  = Wh + (size_t)n * H1 + hh * NHID;
    const float* __restrict__ ah = a + hh * 2 * NHID;
    float r = 0.f, cv = 0.f;
#pragma unroll
    for (int jj = 0; jj < NHID; ++jj) {
        const float v = w[jj];
        r  += v * ah[jj];
        cv += v * ah[NHID + jj];
    }
    sr[t] = r;
    sc[t] = cv;
}

// ---- per-node attention scores, layer 2: sr/sc[N] ------------------------
__global__ void k_scores2(const float* __restrict__ Wh2, const float* __restrict__ aout,
                          float* __restrict__ sr, float* __restrict__ sc, int N) {
    int n = blockIdx.x * blockDim.x + threadIdx.x;
    if (n >= N) return;
    float r = 0.f, cv = 0.f;
#pragma unroll
    for (int j = 0; j < NCLASS; ++j) {
        const float v = Wh2[(size_t)n * NCLASS + j];
        r  += v * aout[j];
        cv += v * aout[NCLASS + j];
    }
    sr[n] = r;
    sc[n] = cv;
}

// ---- layer-1 edge passes --------------------------------------------------
__global__ void k_edge1_max(const int* __restrict__ row, const int* __restrict__ col,
                            const float* __restrict__ sr, const float* __restrict__ sc,
                            unsigned* __restrict__ mk, long long E) {
    long long t = (long long)blockIdx.x * blockDim.x + threadIdx.x;
    if (t >= E) return;
    const int r = row[t], c = col[t];
    const float4 A = *(const float4*)(sr + 4 * (size_t)r);
    const float4 B = *(const float4*)(sc + 4 * (size_t)c);
    unsigned* __restrict__ m = mk + 4 * (size_t)r;
    atomicMax(m + 0, fkey(lrelu(A.x + B.x)));
    atomicMax(m + 1, fkey(lrelu(A.y + B.y)));
    atomicMax(m + 2, fkey(lrelu(A.z + B.z)));
    atomicMax(m + 3, fkey(lrelu(A.w + B.w)));
}

__global__ void k_edge1_sum(const int* __restrict__ row, const int* __restrict__ col,
                            const float* __restrict__ sr, const float* __restrict__ sc,
                            const unsigned* __restrict__ mk, float* __restrict__ ssum,
                            long long E) {
    long long t = (long long)blockIdx.x * blockDim.x + threadIdx.x;
    if (t >= E) return;
    const int r = row[t], c = col[t];
    const float4 A = *(const float4*)(sr + 4 * (size_t)r);
    const float4 B = *(const float4*)(sc + 4 * (size_t)c);
    const uint4  M = *(const uint4*)(mk + 4 * (size_t)r);
    float* __restrict__ s = ssum + 4 * (size_t)r;
    atomicAdd(s + 0, expf(lrelu(A.x + B.x) - keyf(M.x)));
    atomicAdd(s + 1, expf(lrelu(A.y + B.y) - keyf(M.y)));
    atomicAdd(s + 2, expf(lrelu(A.z + B.z) - keyf(M.z)));
    atomicAdd(s + 3, expf(lrelu(A.w + B.w) - keyf(M.w)));
}

// 64 threads per edge, each handles 4 consecutive floats of the 256-wide row
__global__ void k_edge1_agg(const int* __restrict__ row, const int* __restrict__ col,
                            const float* __restrict__ sr, const float* __restrict__ sc,
                            const unsigned* __restrict__ mk, const float* __restrict__ ssum,
                            const float* __restrict__ Wh, float* __restrict__ agg,
                            long long E) {
    long long t = (long long)blockIdx.x * blockDim.x + threadIdx.x;
    const long long e = t >> 6;
    if (e >= E) return;
    const int j  = (int)(t & 63);
    const int hh = j >> 4;          // head
    const int f0 = j * 4;           // feature offset within 256
    const int r = row[e], c = col[e];
    const float v   = lrelu(sr[4 * (size_t)r + hh] + sc[4 * (size_t)c + hh]);
    const float att = expf(v - keyf(mk[4 * (size_t)r + hh])) / ssum[4 * (size_t)r + hh];
    const float4 w  = *(const float4*)(Wh + (size_t)c * H1 + f0);
    float* __restrict__ o = agg + (size_t)r * H1 + f0;
    atomicAdd(o + 0, att * w.x);
    atomicAdd(o + 1, att * w.y);
    atomicAdd(o + 2, att * w.z);
    atomicAdd(o + 3, att * w.w);
}

// ---- elementwise ELU (in place) ------------------------------------------
__global__ void k_elu(float* __restrict__ p, long long n) {
    long long t = (long long)blockIdx.x * blockDim.x + threadIdx.x;
    if (t < n) {
        const float v = p[t];
        p[t] = v > 0.f ? v : expm1f(v);
    }
}

// ---- layer-2 edge passes (single head, 40 features) ----------------------
__global__ void k_edge2_max(const int* __restrict__ row, const int* __restrict__ col,
                            const float* __restrict__ sr, const float* __restrict__ sc,
                            unsigned* __restrict__ mk, long long E) {
    long long t = (long long)blockIdx.x * blockDim.x + threadIdx.x;
    if (t >= E) return;
    const int r = row[t], c = col[t];
    atomicMax(mk + r, fkey(lrelu(sr[r] + sc[c])));
}

__global__ void k_edge2_sum(const int* __restrict__ row, const int* __restrict__ col,
                            const float* __restrict__ sr, const float* __restrict__ sc,
                            const unsigned* __restrict__ mk, float* __restrict__ ssum,
                            long long E) {
    long long t = (long long)blockIdx.x * blockDim.x + threadIdx.x;
    if (t >= E) return;
    const int r = row[t], c = col[t];
    atomicAdd(ssum + r, expf(lrelu(sr[r] + sc[c]) - keyf(mk[r])));
}

// 10 threads per edge, float4 each (40 = 10*4)
__global__ void k_edge2_agg(const int* __restrict__ row, const int* __restrict__ col,
                            const float* __restrict__ sr, const float* __restrict__ sc,
                            const unsigned* __restrict__ mk, const float* __restrict__ ssum,
                            const float* __restrict__ Wh2, float* __restrict__ agg,
                            long long E) {
    long long t = (long long)blockIdx.x * blockDim.x + threadIdx.x;
    const long long e = t / 10;
    if (e >= E) return;
    const int j  = (int)(t - e * 10);
    const int f0 = j * 4;
    const int r = row[e], c = col[e];
    const float v   = lrelu(sr[r] + sc[c]);
    const float att = expf(v - keyf(mk[r])) / ssum[r];
    const float4 w  = *(const float4*)(Wh2 + (size_t)c * NCLASS + f0);
    float* __restrict__ o = agg + (size_t)r * NCLASS + f0;
    atomicAdd(o + 0, att * w.x);
    atomicAdd(o + 1, att * w.y);
    atomicAdd(o + 2, att * w.z);
    atomicAdd(o + 3, att * w.w);
}

// ---- final: ELU + log_softmax per node -----------------------------------
__global__ void k_final(const float* __restrict__ agg2, float* __restrict__ out, int N) {
    int n = blockIdx.x * blockDim.x + threadIdx.x;
    if (n >= N) return;
    const float* __restrict__ a = agg2 + (size_t)n * NCLASS;
    float vals[NCLASS];
    float mx = -INFINITY;
#pragma unroll
    for (int c = 0; c < NCLASS; ++c) {
        float v = a[c];
        v = v > 0.f ? v : expm1f(v);
        vals[c] = v;
        mx = fmaxf(mx, v);
    }
    float s = 0.f;
#pragma unroll
    for (int c = 0; c < NCLASS; ++c) s += expf(vals[c] - mx);
    const float ls = logf(s);
    float* __restrict__ o = out + (size_t)n * NCLASS;
#pragma unroll
    for (int c = 0; c < NCLASS; ++c) o[c] = vals[c] - mx - ls;
}

extern "C" void kernel_launch(void* const* d_in, const int* in_sizes, int n_in,
                              void* d_out, int out_size, void* d_ws, size_t ws_size,
                              hipStream_t stream) {
    const float* x    = (const float*)d_in[0];
    const float* W    = (const float*)d_in[1];
    const float* a    = (const float*)d_in[2];
    const float* Wout = (const float*)d_in[3];
    const float* aout = (const float*)d_in[4];
    const int*   row  = (const int*)d_in[5];
    const int*   col  = (const int*)d_in[6];

    const int       N = in_sizes[0] / NFEAT;   // 100000 (divisible by 16)
    const long long E = in_sizes[5];           // 1,700,000

    // ---- workspace layout (floats); zero-init region is contiguous ----
    float* ws = (float*)d_ws;
    float*    Wh1  = ws;              ws += (size_t)N * H1;
    float*    sr1  = ws;              ws += (size_t)N * NHEADS;
    float*    sc1  = ws;              ws += (size_t)N * NHEADS;
    float*    Wh2  = ws;              ws += (size_t)N * NCLASS;
    float*    sr2  = ws;              ws += N;
    float*    sc2  = ws;              ws += N;
    float*    zbase = ws;
    float*    agg1 = ws;              ws += (size_t)N * H1;
    unsigned* m1   = (unsigned*)ws;   ws += (size_t)N * NHEADS;
    float*    ss1  = ws;              ws += (size_t)N * NHEADS;
    unsigned* m2   = (unsigned*)ws;   ws += N;
    float*    ss2  = ws;              ws += N;
    float*    agg2 = ws;              ws += (size_t)N * NCLASS;
    const long long zcount = (long long)(ws - zbase);

    const int B = 256;
    k_zero<<<(unsigned)((zcount + B - 1) / B), B, 0, stream>>>((unsigned*)zbase, zcount);

    // Layer 1
    dim3 g1(N / 16, 4);
    k_gemm1<<<g1, 128, 0, stream>>>(x, W, Wh1, N);
    k_scores1<<<(unsigned)(((long long)N * NHEADS + B - 1) / B), B, 0, stream>>>(Wh1, a, sr1, sc1, N);
    k_edge1_max<<<(unsigned)((E + B - 1) / B), B, 0, stream>>>(row, col, sr1, sc1, m1, E);
    k_edge1_sum<<<(unsigned)((E + B - 1) / B), B, 0, stream>>>(row, col, sr1, sc1, m1, ss1, E);
    k_edge1_agg<<<(unsigned)((E * 64 + B - 1) / B), B, 0, stream>>>(row, col, sr1, sc1, m1, ss1, Wh1, agg1, E);
    k_elu<<<(unsigned)(((long long)N * H1 + B - 1) / B), B, 0, stream>>>(agg1, (long long)N * H1);

    // Layer 2
    dim3 g2(N / 16, 3);
    k_gemm2<<<g2, 32, 0, stream>>>(agg1, Wout, Wh2, N);
    k_scores2<<<(unsigned)((N + B - 1) / B), B, 0, stream>>>(Wh2, aout, sr2, sc2, N);
    k_edge2_max<<<(unsigned)((E + B - 1) / B), B, 0, stream>>>(row, col, sr2, sc2, m2, E);
    k_edge2_sum<<<(unsigned)((E + B - 1) / B), B, 0, stream>>>(row, col, sr2, sc2, m2, ss2, E);
    k_edge2_agg<<<(unsigned)((E * 10 + B - 1) / B), B, 0, stream>>>(row, col, sr2, sc2, m2, ss2, Wh2, agg2, E);
    k_final<<<(unsigned)((N + B - 1) / B), B, 0, stream>>>(agg2, (float*)d_out, N);
}